// Example1_2954937500140
// MI455X (gfx1250) — compile-verified
//
#include <hip/hip_runtime.h>
#include <math.h>

#define VOCAB 2048
#define CTX   1024
#define DDIM  3072
#define NTHREADS 1024

typedef __attribute__((ext_vector_type(2))) float v2f;
typedef __attribute__((ext_vector_type(8))) float v8f;

// One workgroup per batch. Exploits the one-hot structure of enc:
//   encR_last[e] = R[t_last, e] + R[3071, e]            (LDS vector v)
//   scores[m]    = v[tok[m]] + v[2048 + m]
//   A            = softmax(scores)                       (sum via v_wmma_f32_16x16x4_f32, exact fp32)
//   out[t<2048]  = sum_{m: tok[m]==t} A[m]   (deterministic gather, no float atomics)
//   out[2048+m]  = A[m]
__global__ __launch_bounds__(NTHREADS)
void lastrow_attn_kernel(const int* __restrict__ token_ids,
                         const float* __restrict__ R,
                         float* __restrict__ out)
{
    __shared__ __align__(16) float v[DDIM];   // encR_last
    __shared__ int   tok[CTX];
    __shared__ __align__(16) float ev[CTX];   // exp(scores - max)
    __shared__ float red[32];
    __shared__ float bc[2];                   // [0]=max, [1]=sum

    const int b    = blockIdx.x;
    const int tid  = threadIdx.x;
    const int lane = tid & 31;
    const int wid  = tid >> 5;

    const int* tk_b = token_ids + b * CTX;
    tok[tid] = tk_b[tid];
    const int t_last = tk_b[CTX - 1];

    // encR_last = R[t_last, :] + R[VOCAB+CTX-1, :]
    const float* Rt = R + (size_t)t_last * DDIM;
    const float* Rp = R + (size_t)(VOCAB + CTX - 1) * DDIM;
    for (int e = tid; e < DDIM; e += NTHREADS)
        v[e] = Rt[e] + Rp[e];
    __syncthreads();

    // score for m = tid (causal mask vacuous for the last row)
    const float s = v[tok[tid]] + v[VOCAB + tid];

    // ---- max reduction (wave shfl + LDS combine) ----
    float mx = s;
#pragma unroll
    for (int off = 16; off > 0; off >>= 1)
        mx = fmaxf(mx, __shfl_xor(mx, off, 32));
    if (lane == 0) red[wid] = mx;
    __syncthreads();
    if (wid == 0) {
        float t = red[lane];
#pragma unroll
        for (int off = 16; off > 0; off >>= 1)
            t = fmaxf(t, __shfl_xor(t, off, 32));
        if (lane == 0) bc[0] = t;
    }
    __syncthreads();
    const float gmax = bc[0];

    const float e = expf(s - gmax);
    ev[tid] = e;
    __syncthreads();

    // ---- sum of 1024 fp32 values via WMMA (wave 0 only; EXEC all-ones) ----
    // A = 64 data values per step, B = ones(4x16) => D[i][j] accumulates rowsum_i(A).
    // Sum of all 256 D elements = 16 * total; multiply by 1/16 (exact in fp32).
    if (wid == 0) {
        v2f ones;
        ones.x = 1.0f;
        ones.y = 1.0f;
        v8f c = {};
#pragma unroll
        for (int k = 0; k < 16; ++k) {
            v2f a = *(const v2f*)&ev[k * 64 + lane * 2];
            c = __builtin_amdgcn_wmma_f32_16x16x4_f32(
                    /*neg_a=*/false, a, /*neg_b=*/false, ones,
                    /*c_mod=*/(short)0, c, /*reuse_a=*/false, /*reuse_b=*/false);
        }
        float p = c[0] + c[1] + c[2] + c[3] + c[4] + c[5] + c[6] + c[7];
#pragma unroll
        for (int off = 16; off > 0; off >>= 1)
            p += __shfl_xor(p, off, 32);
        if (lane == 0) bc[1] = p * 0.0625f;   // /16, exact
    }
    __syncthreads();
    const float inv_total = 1.0f / bc[1];

    float* ob = out + (size_t)b * DDIM;

    // positional part of logits
    ob[VOCAB + tid] = e * inv_total;

    // token part: deterministic scatter-as-gather (LDS broadcast reads)
    float acc0 = 0.0f, acc1 = 0.0f;
    const int t0 = tid;
    const int t1 = tid + NTHREADS;            // covers 0..2047
    for (int m = 0; m < CTX; ++m) {
        const float a  = ev[m];
        const int  tkm = tok[m];
        if (tkm == t0) acc0 += a;
        if (tkm == t1) acc1 += a;
    }
    ob[t0] = acc0 * inv_total;
    ob[t1] = acc1 * inv_total;
}

extern "C" void kernel_launch(void* const* d_in, const int* in_sizes, int n_in,
                              void* d_out, int out_size, void* d_ws, size_t ws_size,
                              hipStream_t stream) {
    const int*   token_ids = (const int*)d_in[0];    // 32 x 1024
    const float* R         = (const float*)d_in[1];  // 3072 x 3072
    float*       out       = (float*)d_out;          // 32 x 3072

    (void)in_sizes; (void)n_in; (void)out_size; (void)d_ws; (void)ws_size;

    lastrow_attn_kernel<<<32, NTHREADS, 0, stream>>>(token_ids, R, out);
}